// CRF_Cell_3435973837484
// MI455X (gfx1250) — compile-verified
//
#include <hip/hip_runtime.h>
#include <hip/hip_bf16.h>

// ---------------------------------------------------------------------------
// CRF loss (all_path - truth_path) for 8192 independent length-128 sequences.
//  Kernel A: per-thread sequential forward recurrence (1 seq / thread),
//            features + ids staged to LDS with CDNA5 async-to-LDS DMA
//            (global_load_async_to_lds_b128, s_wait_asynccnt), double buffered.
//  Kernel T: truth-path parallel gather/sum, per-block partials (negated).
//  Kernel R: deterministic final reduction via V_WMMA_F32_16X16X4_F32
//            (A = ones -> column sums; layout-independent exact sum).
// ---------------------------------------------------------------------------

#define TAGN          8
#define START_ID_C    101
#define STOP_ID_C     102
#define SEQ_LEN       128
#define SEQ_PER_BLOCK 64
#define BLOCK_A       64
#define CHUNK         8                       // positions per LDS tile
#define NT            (SEQ_LEN / CHUNK)       // 16 tiles
#define FEAT_SEQ_STRIDE_B 272                 // CHUNK*8*4 + 16B pad (bank spread)
#define FEAT_BUF_B   (SEQ_PER_BLOCK * FEAT_SEQ_STRIDE_B)   // 17408 B
#define IDS_SEQ_STRIDE_B  32                  // CHUNK*4
#define IDS_BUF_B    (SEQ_PER_BLOCK * IDS_SEQ_STRIDE_B)    // 2048 B
#define NB_T          256                     // truth-kernel blocks

typedef __attribute__((ext_vector_type(2))) float v2f;
typedef __attribute__((ext_vector_type(8))) float v8f;

__device__ __forceinline__ unsigned lds_addr32(const void* p) {
  // flat LDS pointer: low 32 bits == LDS byte offset (aperture in high bits)
  return (unsigned)(unsigned long long)p;
}

__device__ __forceinline__ void async_b128(unsigned lds, unsigned long long gaddr) {
  // GLOBAL_LOAD_ASYNC_TO_LDS_B128: VDST = per-lane LDS byte address,
  // VADDR = per-lane 64-bit global address. Tracked by ASYNCcnt.
  asm volatile("global_load_async_to_lds_b128 %0, %1, off"
               :: "v"(lds), "v"(gaddr) : "memory");
}

// Stage one tile: features [seq0..seq0+63][pos0..pos0+7][8 f32] and ids.
// 16 feat issues + 2 id issues per wave => ASYNCcnt += 18 per fill.
__device__ __forceinline__ void fill_tile(const float* wf, const int* ids,
                                          int seq0, int pos0, int buf,
                                          unsigned featBase, unsigned idsBase,
                                          int tid) {
  unsigned long long wfb =
      (unsigned long long)(const void*)wf + (unsigned long long)seq0 * SEQ_LEN * 32ull;
  #pragma unroll
  for (int q = 0; q < 16; ++q) {
    int c = q * BLOCK_A + tid;               // 0..1023 16B chunks
    int seq = c >> 4, k = c & 15;            // 16 chunks per sequence row
    unsigned long long g = wfb + (unsigned long long)seq * (SEQ_LEN * 32)
                         + (unsigned)(pos0 * 32 + k * 16);
    unsigned l = featBase + (unsigned)(buf * FEAT_BUF_B + seq * FEAT_SEQ_STRIDE_B + k * 16);
    async_b128(l, g);
  }
  unsigned long long idb =
      (unsigned long long)(const void*)ids + (unsigned long long)seq0 * SEQ_LEN * 4ull;
  #pragma unroll
  for (int q = 0; q < 2; ++q) {
    int c = q * BLOCK_A + tid;               // 0..127 16B chunks
    int seq = c >> 1, k = c & 1;             // 2 chunks per sequence row
    unsigned long long g = idb + (unsigned long long)seq * (SEQ_LEN * 4)
                         + (unsigned)(pos0 * 4 + k * 16);
    unsigned l = idsBase + (unsigned)(buf * IDS_BUF_B + seq * IDS_SEQ_STRIDE_B + k * 16);
    async_b128(l, g);
  }
}

__global__ void __launch_bounds__(BLOCK_A)
allpath_kernel(const float* __restrict__ wf, const int* __restrict__ ids,
               const float* __restrict__ tr, float* __restrict__ partA) {
  __shared__ __align__(16) char sFeat[2 * FEAT_BUF_B];
  __shared__ __align__(16) char sIds [2 * IDS_BUF_B];
  __shared__ float sRed[BLOCK_A];

  const int tid  = threadIdx.x;
  const int seq0 = blockIdx.x * SEQ_PER_BLOCK;
  const unsigned featBase = lds_addr32(sFeat);
  const unsigned idsBase  = lds_addr32(sIds);

  // Tt[i][j] = transitions[j][i]; uniform address -> scalar loads.
  float tt[8][8];
  #pragma unroll
  for (int i = 0; i < 8; ++i)
    #pragma unroll
    for (int j = 0; j < 8; ++j) tt[i][j] = tr[j * 8 + i];

  // Prime double buffer.
  fill_tile(wf, ids, seq0, 0,     0, featBase, idsBase, tid);
  fill_tile(wf, ids, seq0, CHUNK, 1, featBase, idsBase, tid);

  float fv[8];
  #pragma unroll
  for (int i = 0; i < 8; ++i) fv[i] = 0.f;
  float acc = 0.f;

  for (int t = 0; t < NT; ++t) {
    // One later fill (18 async instrs) may be outstanding; in-order completion
    // means ASYNCcnt<=18 guarantees tile t is resident (<=0 for the last tile).
    if (t + 1 < NT) asm volatile("s_wait_asynccnt 18" ::: "memory");
    else            asm volatile("s_wait_asynccnt 0"  ::: "memory");
    __syncthreads();

    const int buf = t & 1;
    const float* fRow = (const float*)(sFeat + buf * FEAT_BUF_B + tid * FEAT_SEQ_STRIDE_B);
    const int*   iRow = (const int*)  (sIds  + buf * IDS_BUF_B  + tid * IDS_SEQ_STRIDE_B);

    for (int p = 0; p < CHUNK; ++p) {
      const int tok = iRow[p];
      float f[8];
      #pragma unroll
      for (int j = 0; j < 8; ++j) f[j] = fRow[p * 8 + j];

      if (tok == STOP_ID_C) {
        // exact logsumexp(fv); fv unchanged (scan keeps fv at STOP)
        float m = fv[0];
        #pragma unroll
        for (int i = 1; i < 8; ++i) m = fmaxf(m, fv[i]);
        float s = 0.f;
        #pragma unroll
        for (int i = 0; i < 8; ++i) s += __expf(fv[i] - m);
        acc += m + __logf(s);
      } else if (tok == START_ID_C) {
        #pragma unroll
        for (int i = 0; i < 8; ++i) fv[i] = 0.f;
      } else {
        // buggy LSE: score[i][j] = fv[i] + f[j] + tt[i][j]
        float r0[8];
        #pragma unroll
        for (int j = 0; j < 8; ++j) r0[j] = f[j] + tt[0][j];
        float msc[8];
        float S = 0.f;
        #pragma unroll
        for (int i = 0; i < 8; ++i) {
          float bi[8];
          bi[0] = f[0] + tt[i][0];
          float best = bi[0];
          int   idx  = 0;
          #pragma unroll
          for (int j = 1; j < 8; ++j) {          // first-max wins (strict >)
            bi[j] = f[j] + tt[i][j];
            if (bi[j] > best) { best = bi[j]; idx = j; }
          }
          float r = r0[0];                       // gather score[0, idx]
          #pragma unroll
          for (int k = 1; k < 8; ++k) r = (idx == k) ? r0[k] : r;
          const float m = fv[0] + r;             // maxsc[i]
          msc[i] = m;
          float se = 0.f;
          #pragma unroll
          for (int j = 0; j < 8; ++j) se += __expf(fv[i] + bi[j] - m);
          S += se;                               // sum over ALL 64 elements
        }
        const float lS = __logf(S);
        #pragma unroll
        for (int i = 0; i < 8; ++i) fv[i] = msc[i] + lS;
      }
    }
    __syncthreads();
    if (t + 2 < NT)
      fill_tile(wf, ids, seq0, (t + 2) * CHUNK, buf, featBase, idsBase, tid);
  }

  sRed[tid] = acc;
  __syncthreads();
  if (tid == 0) {                                 // fixed-order => deterministic
    float s = 0.f;
    for (int i = 0; i < BLOCK_A; ++i) s += sRed[i];
    partA[blockIdx.x] = s;
  }
}

__device__ __forceinline__ int tag_of(int tok, int tc) {
  return tok == START_ID_C ? 6 : (tok == STOP_ID_C ? 7 : tc);
}

__global__ void __launch_bounds__(256)
truth_kernel(const float* __restrict__ wf, const int* __restrict__ ids,
             const int* __restrict__ tcol, const float* __restrict__ tr,
             float* __restrict__ partT, int T) {
  __shared__ float sRed[256];
  float s = 0.f;
  for (int t = blockIdx.x * blockDim.x + threadIdx.x; t < T;
       t += gridDim.x * blockDim.x) {
    const int tag = tag_of(ids[t], tcol[t]);
    if (tag < 6) {                                // interior token
      const int tn = (t + 1 < T) ? t + 1 : 0;     // jnp.roll(-1) wrap
      const int tagn = tag_of(ids[tn], tcol[tn]);
      s += wf[t * 8 + tag] + tr[tagn * 8 + tag];
    } else if (tag == 7) {                        // STOP: minus sign in source
      const int tp = (t > 0) ? t - 1 : T - 1;     // jnp.roll(+1) wrap
      const int tagp = tag_of(ids[tp], tcol[tp]);
      s -= tr[7 * 8 + tagp];
    }
  }
  sRed[threadIdx.x] = s;
  __syncthreads();
  if (threadIdx.x == 0) {
    float b = 0.f;
    for (int i = 0; i < 256; ++i) b += sRed[i];
    partT[blockIdx.x] = -b;                       // negate: total = all - truth
  }
}

// Final reduction on the matrix pipe: D = ones(16x4) x B + C accumulates
// column sums of B; sum of D row 0 == sum of all 64 packed values,
// independent of the exact B lane layout. Exact (each addend once).
// Caller guarantees n is a multiple of 64 (384 = 6 tiles here), so the
// inner loop is branch-free: 2 loads + 1 v_wmma, EXEC all ones throughout.
__global__ void __launch_bounds__(32)
reduce_kernel(const float* __restrict__ parts, int n, float* __restrict__ out) {
  const int lane = threadIdx.x;                   // wave32, EXEC all ones
  v8f acc = {0.f, 0.f, 0.f, 0.f, 0.f, 0.f, 0.f, 0.f};
  v2f ones; ones.x = 1.f; ones.y = 1.f;
  const int tiles = n >> 6;                       // exact: n % 64 == 0
  for (int t = 0; t < tiles; ++t) {
    const int i0 = t * 64 + lane;
    v2f b;
    b.x = parts[i0];
    b.y = parts[i0 + 32];
    acc = __builtin_amdgcn_wmma_f32_16x16x4_f32(
        /*neg_a=*/false, ones, /*neg_b=*/false, b,
        /*c_mod=*/(short)0, acc, /*reuse_a=*/false, /*reuse_b=*/false);
  }
  // D row 0 (M=0) = acc element 0, lanes 0..15 (ISA 16x16 f32 C/D layout).
  const float r = acc[0];
  float tot = 0.f;
  for (int l = 0; l < 16; ++l) tot += __shfl(r, l, 32);
  if (lane == 0) out[0] = tot;
}

extern "C" void kernel_launch(void* const* d_in, const int* in_sizes, int n_in,
                              void* d_out, int out_size, void* d_ws, size_t ws_size,
                              hipStream_t stream) {
  const float* wf   = (const float*)d_in[0];     // [T, 8] f32
  const int*   ids  = (const int*)  d_in[1];     // [T] i32
  const int*   tcol = (const int*)  d_in[2];     // [T] i32
  const float* tr   = (const float*)d_in[3];     // [8, 8] f32
  float*       out  = (float*)d_out;

  const int T    = in_sizes[1];                  // 1048576
  const int nseq = T / SEQ_LEN;                  // 8192
  const int nbA  = nseq / SEQ_PER_BLOCK;         // 128

  float* partA = (float*)d_ws;                   // [nbA]
  float* partT = partA + nbA;                    // [NB_T]

  allpath_kernel<<<nbA, BLOCK_A, 0, stream>>>(wf, ids, tr, partA);
  truth_kernel<<<NB_T, 256, 0, stream>>>(wf, ids, tcol, tr, partT, T);
  // nbA + NB_T = 128 + 256 = 384, a multiple of 64 -> branch-free WMMA reduce.
  reduce_kernel<<<1, 32, 0, stream>>>(partA, nbA + NB_T, out);
}